// Proposal_21878563406368
// MI455X (gfx1250) — compile-verified
//
#include <hip/hip_runtime.h>

// Problem constants (shapes fixed by the harness):
//   real_B/fake_B/real_A: [B,3,512,512] f32, score_map: [B,1,128,128] f32, B=32
//   img_stride = floor((512-70)/128) = 3
#define SCORE_W  128
#define SCORE_N  (SCORE_W * SCORE_W)   // 16384
#define IMG_H    512
#define REGION   70
#define STRIDE_I 3
#define THREADS  256

typedef __attribute__((ext_vector_type(4))) unsigned int u32x4;
typedef __attribute__((ext_vector_type(8))) int i32x8;
typedef __attribute__((ext_vector_type(4))) int i32x4;

#if defined(__gfx1250__) && __has_builtin(__builtin_amdgcn_tensor_load_to_lds) && \
    __has_builtin(__builtin_amdgcn_s_wait_tensorcnt)
#define USE_TDM 1
#else
#define USE_TDM 0
#endif

#if USE_TDM
// Build a 1-row (16384-element, 64KB) f32 tile descriptor and DMA it into LDS
// via the Tensor Data Mover. D# layout per CDNA5 ISA ch.8 (CDNA5 ISA).
// This toolchain (clang-23/therock) uses the 6-arg builtin:
//   (u32x4 g0, i32x8 g1, i32x4 g2, i32x4 g3, i32x8 extra, i32 cpol)
__device__ __forceinline__ void tdm_load_scores(const float* __restrict__ gsrc,
                                                unsigned lds_byte_off) {
    unsigned long long ga = (unsigned long long)(uintptr_t)gsrc;
    u32x4 g0;
    g0[0] = 1u;                                   // count=1 (valid user descriptor)
    g0[1] = lds_byte_off;                         // lds_addr
    g0[2] = (unsigned)(ga & 0xFFFFFFFFull);       // global_addr[31:0]
    g0[3] = (unsigned)((ga >> 32) & 0x01FFFFFFull)// global_addr[56:32]
            | (2u << 30);                         // type = 2 ("image")
    i32x8 g1;
    g1[0] = (int)(2u << 16);                      // data_size=2 (4B), no multicast/pad
    g1[1] = (int)((unsigned)SCORE_N << 16);       // tensor_dim0[15:0] -> bits[63:48]
    g1[2] = (int)(1u << 16);                      // tensor_dim0 hi=0 | tensor_dim1=1
    g1[3] = (int)((unsigned)SCORE_N << 16);       // tensor_dim1 hi=0 | tile_dim0=16384
    g1[4] = 1;                                    // tile_dim1=1, tile_dim2=0
    g1[5] = SCORE_N;                              // tensor_dim0_stride[31:0]
    g1[6] = (int)((unsigned)SCORE_N << 16);       // stride0 hi=0 | stride1 lo16
    g1[7] = 0;                                    // stride1 hi
    i32x4 z4 = {0, 0, 0, 0};                      // groups 2/3 unused (<=2D tensor)
    i32x8 z8 = {0, 0, 0, 0, 0, 0, 0, 0};
    __builtin_amdgcn_tensor_load_to_lds(g0, g1, z4, z4, z8, 0);
    __builtin_amdgcn_s_wait_tensorcnt(0);
}
#endif

// One block per batch image. Stage 64KB score tile into LDS (TDM), then
// first-occurrence argmax/argmin reduction, then emit integer box corners:
// boxes[b] = {row_real, col_real, row_fake, col_fake}.
__global__ void localize_kernel(const float* __restrict__ score,
                                int* __restrict__ boxes) {
    __shared__ __align__(16) unsigned char smem[SCORE_N * 4];  // 64KB
    float* tile = (float*)smem;
    const int b = blockIdx.x;
    const int t = threadIdx.x;
    const float* src = score + (size_t)b * SCORE_N;

#if USE_TDM
    if (t == 0) {
        tdm_load_scores(src, 0u);   // TDM issue + s_wait_tensorcnt 0 (wave 0 only)
    }
#else
    for (int k = t; k < SCORE_N; k += THREADS) tile[k] = src[k];
#endif
    __syncthreads();

    // Strided scan: idx = k*256 + t  -> LDS-bank-conflict-free, and per-thread
    // indices are strictly increasing, so strict compares keep first occurrence.
    float vmax = -3.402823466e+38f; int imax = 0;
    float vmin =  3.402823466e+38f; int imin = 0;
    #pragma unroll 4
    for (int k = 0; k < SCORE_N / THREADS; ++k) {
        const int idx = (k << 8) + t;
        const float v = tile[idx];
        if (v > vmax) { vmax = v; imax = idx; }
        if (v < vmin) { vmin = v; imin = idx; }
    }
    __syncthreads();   // done reading tile; reuse its first 4KB for reduction

    float* rMaxV = (float*)(smem);
    int*   rMaxI = (int*)  (smem + 1024);
    float* rMinV = (float*)(smem + 2048);
    int*   rMinI = (int*)  (smem + 3072);
    rMaxV[t] = vmax; rMaxI[t] = imax;
    rMinV[t] = vmin; rMinI[t] = imin;
    __syncthreads();

    // Tree reduction with lower-index tie-break (first-occurrence semantics).
    for (int s = THREADS / 2; s > 0; s >>= 1) {
        if (t < s) {
            float bv = rMaxV[t + s]; int bi = rMaxI[t + s];
            if (bv > rMaxV[t] || (bv == rMaxV[t] && bi < rMaxI[t])) {
                rMaxV[t] = bv; rMaxI[t] = bi;
            }
            float cv = rMinV[t + s]; int ci = rMinI[t + s];
            if (cv < rMinV[t] || (cv == rMinV[t] && ci < rMinI[t])) {
                rMinV[t] = cv; rMinI[t] = ci;
            }
        }
        __syncthreads();
    }

    if (t == 0) {
        const float vM = rMaxV[0]; const int iM = rMaxI[0];
        const float vm = rMinV[0]; const int im = rMinI[0];
        int rr, cr, rf, cf;
        if (vM > 0.0f) { rr = (iM >> 7) * STRIDE_I + REGION;
                         cr = (iM & (SCORE_W - 1)) * STRIDE_I + REGION; }
        else           { rr = REGION; cr = REGION; }              // coord 0
        if (vm < 1.0f) { rf = (im >> 7) * STRIDE_I + REGION;
                         cf = (im & (SCORE_W - 1)) * STRIDE_I + REGION; }
        else           { rf = STRIDE_I + REGION; cf = STRIDE_I + REGION; } // coord 1
        int4 bx; bx.x = rr; bx.y = cr; bx.z = rf; bx.w = cf;
        ((int4*)boxes)[b] = bx;
    }
}

// grid = (B*4, 3).  blockIdx.x encodes (batch, which):
//   which = 0: fake_B @ real box -> out0,           real_ABr chans 3..5
//   which = 1: real_A @ real box -> out1,           real_ABr chans 0..2
//   which = 2: fake_B @ fake box -> out2,           fake_ABf chans 3..5
//   which = 3: real_A @ fake box -> out3,           fake_ABf chans 0..2
// blockIdx.y = channel. Pure masked gather (sample positions are exact ints).
__global__ void crop_kernel(const float* __restrict__ fake_B,
                            const float* __restrict__ real_A,
                            const int* __restrict__ boxes,
                            float* __restrict__ out, int B) {
    const int bid   = blockIdx.x;
    const int b     = bid >> 2;
    const int which = bid & 3;
    const int c     = blockIdx.y;

    const float* src = (which & 1) ? real_A : fake_B;
    const int* bx = boxes + b * 4;
    const int row0 = (which < 2) ? bx[0] : bx[2];
    const int col0 = (which < 2) ? bx[1] : bx[3];

    const long long crop  = (long long)REGION * REGION;      // 4900
    const long long perB3 = 3 * crop;                        // 14700
    const long long perB6 = 6 * crop;                        // 29400
    const long long O4    = 4LL * B * perB3;                 // fake_ABf base
    const long long O5    = O4 + (long long)B * perB6;       // real_ABr base

    const long long Oab   = (which < 2) ? O5 : O4;
    const int chOff       = (which & 1) ? 0 : 3;

    float* dst0  = out + (long long)which * B * perB3 + (long long)b * perB3
                       + (long long)c * crop;
    float* dstAB = out + Oab + (long long)b * perB6
                       + (long long)(c + chOff) * crop;
    const float* sb = src + ((size_t)b * 3 + c) * IMG_H * IMG_H;

    for (int e = threadIdx.x; e < (int)crop; e += THREADS) {
        const int i = e / REGION;
        const int j = e - i * REGION;
        const int r  = row0 + i;
        const int cc = col0 + j;
        float v = 0.0f;
        if (r < IMG_H && cc < IMG_H)            // row0,col0 >= 70, no lower check
            v = sb[((size_t)r << 9) + cc];
        dst0[e]  = v;
        dstAB[e] = v;
    }
}

extern "C" void kernel_launch(void* const* d_in, const int* in_sizes, int n_in,
                              void* d_out, int out_size, void* d_ws, size_t ws_size,
                              hipStream_t stream) {
    // setup_inputs order: real_B (unused), fake_B, real_A, score_map
    const float* fake_B = (const float*)d_in[1];
    const float* real_A = (const float*)d_in[2];
    const float* score  = (const float*)d_in[3];
    float* out = (float*)d_out;
    int* boxes = (int*)d_ws;                       // B * 4 ints
    const int B = in_sizes[3] / SCORE_N;           // 32

    localize_kernel<<<dim3(B), dim3(THREADS), 0, stream>>>(score, boxes);
    crop_kernel<<<dim3(B * 4, 3), dim3(THREADS), 0, stream>>>(fake_B, real_A,
                                                              boxes, out, B);
}